// LogEig_49752901157019
// MI455X (gfx1250) — compile-verified
//
#include <hip/hip_runtime.h>

// fp32 WMMA fragment types (wave32):
//  A 16x4  f32 -> 2 VGPRs/lane, B 4x16 f32 -> 2 VGPRs/lane, C/D 16x16 f32 -> 8 VGPRs/lane
typedef float f32x2 __attribute__((ext_vector_type(2)));
typedef float f32x8 __attribute__((ext_vector_type(8)));

#define LDSTRIDE 68  // 68 % 64 banks = 4-bank shift/row -> conflict-free column reads

// 64x64 fp32 GEMM on one workgroup (8 wave32), via v_wmma_f32_16x16x4_f32.
// Each wave computes two 16x16 tiles: row block = wave&3, col blocks = (wave>>2)*32 + {0,16}.
// k-loop is outer so the A fragment (shared by both tiles) is loaded once per k-step,
// as an aligned 8-byte LDS vector load.
// MODE 0: dst = A*B                      (LDS, stride LDSTRIDE)
// MODE 1: dst = 1.5*I - 0.5*(A*B)        (Newton-Schulz helper T)
// MODE 2: dst = coef*I + (A*B)           (Horner step)
// MODE 3: dst = coef*(A*B) + coef2*I     (global store, stride 64)
template <int MODE, bool SYNC>
__device__ __forceinline__ void mm64(float* __restrict__ dst,
                                     const float* __restrict__ A,
                                     const float* __restrict__ B,
                                     float coef, float coef2) {
  const int tid  = threadIdx.x;
  const int wave = tid >> 5;
  const int lane = tid & 31;
  const int half = lane >> 4;   // 0 = lanes 0-15, 1 = lanes 16-31
  const int l16  = lane & 15;
  const int rb   = (wave & 3) * 16;        // C row block
  const int cb0  = (wave >> 2) * 32;       // first of two C col blocks
  const int cb1  = cb0 + 16;

  f32x8 acc0 = {0.f, 0.f, 0.f, 0.f, 0.f, 0.f, 0.f, 0.f};
  f32x8 acc1 = {0.f, 0.f, 0.f, 0.f, 0.f, 0.f, 0.f, 0.f};

#pragma unroll
  for (int k0 = 0; k0 < 64; k0 += 4) {
    // A frag (16x4): lane row = rb+l16; VGPR0 = K(k0+2*half), VGPR1 = K(k0+2*half+1).
    // Address is always 8B-aligned (even dword) -> single ds_load_b64 into a pair.
    const f32x2 av = *(const f32x2*)(A + (rb + l16) * LDSTRIDE + (k0 + 2 * half));
    // B frags (4x16): lane col = cb+l16; VGPR0 = row K(k0+2*half), VGPR1 = row +1.
    const float* bp = B + (k0 + 2 * half) * LDSTRIDE + l16;
    f32x2 b0, b1;
    b0.x = bp[cb0];
    b0.y = bp[cb0 + LDSTRIDE];
    b1.x = bp[cb1];
    b1.y = bp[cb1 + LDSTRIDE];
    acc0 = __builtin_amdgcn_wmma_f32_16x16x4_f32(
        false, av, false, b0, (short)0, acc0, false, false);
    acc1 = __builtin_amdgcn_wmma_f32_16x16x4_f32(
        false, av, false, b1, (short)0, acc1, false, false);
  }

  // C/D layout: VGPR j -> row rb + j + 8*half, col cb + l16
#pragma unroll
  for (int t = 0; t < 2; ++t) {
    const int cb = t ? cb1 : cb0;
    const f32x8& acc = t ? acc1 : acc0;
#pragma unroll
    for (int j = 0; j < 8; ++j) {
      const int r = rb + j + 8 * half;
      const int c = cb + l16;
      float v = acc[j];
      if (MODE == 1) v = ((r == c) ? 1.5f : 0.0f) - 0.5f * v;
      if (MODE == 2) v = v + ((r == c) ? coef : 0.0f);
      if (MODE == 3) {
        dst[r * 64 + c] = coef * v + ((r == c) ? coef2 : 0.0f);
      } else {
        dst[r * LDSTRIDE + c] = v;
      }
    }
  }
  if (SYNC) __syncthreads();
}

// log(X) per 64x64 SPD matrix: inverse scaling-and-squaring.
//   c = ||X||_F (>= lambda_max)  ->  Xn = X/c has spectrum in (0,1]
//   6 coupled Newton-Schulz square roots: M = Xn^(1/64), ||M - I|| <~ 0.14
//   log(Xn) ~= 64 * Mercator7(M - I)   (Horner, all matmuls)
//   log(X)  = log(Xn) + log(c)*I
__global__ __launch_bounds__(256) void logeig64_kernel(const float* __restrict__ Xg,
                                                       float* __restrict__ Og) {
  __shared__ float buf[5][64 * LDSTRIDE];
  __shared__ float red[256];
  __shared__ float scal[2];  // [0] = 1/c, [1] = log(c)

  const int tid = threadIdx.x;
  const float* X = Xg + (size_t)blockIdx.x * 4096;
  float* O = Og + (size_t)blockIdx.x * 4096;

  float* Y  = buf[0];
  float* Z  = buf[1];
  float* T  = buf[2];
  float* Y2 = buf[3];
  float* Z2 = buf[4];

  // Load X into Y, accumulate Frobenius norm.
  float ss = 0.f;
  for (int i = tid; i < 4096; i += 256) {
    const int r = i >> 6, c = i & 63;
    const float v = X[i];
    Y[r * LDSTRIDE + c] = v;
    ss += v * v;
  }
  red[tid] = ss;
  __syncthreads();
#pragma unroll
  for (int off = 128; off > 0; off >>= 1) {
    if (tid < off) red[tid] += red[tid + off];
    __syncthreads();
  }
  if (tid == 0) {
    float cn = fmaxf(sqrtf(red[0]), 1e-30f);
    scal[0] = 1.0f / cn;
    scal[1] = logf(cn);
  }
  __syncthreads();

  const float rcp = scal[0];
  for (int i = tid; i < 4096; i += 256) {
    const int r = i >> 6, c = i & 63;
    Y[r * LDSTRIDE + c] *= rcp;                      // Y = X / c
    Z[r * LDSTRIDE + c] = (r == c) ? 1.0f : 0.0f;    // Z = I
  }
  __syncthreads();

  // 6 successive square roots via coupled Newton-Schulz (Y -> M^(1/2), Z -> M^(-1/2)).
#pragma unroll 1
  for (int s = 0; s < 6; ++s) {
    if (s > 0) {
      for (int i = tid; i < 4096; i += 256) {
        const int r = i >> 6, c = i & 63;
        Z[r * LDSTRIDE + c] = (r == c) ? 1.0f : 0.0f;
      }
      __syncthreads();
    }
    // Spectrum tightens each sqrt -> taper the iteration count.
    const int ni = (s == 0) ? 18 : (s == 1) ? 14 : (s == 2) ? 11
                 : (s == 3) ? 9  : (s == 4) ? 8  : 7;
#pragma unroll 1
    for (int it = 0; it < ni; ++it) {
      mm64<1, true >(T,  Z, Y, 0.f, 0.f);   // T  = 1.5I - 0.5*(Z*Y)
      mm64<0, false>(Y2, Y, T, 0.f, 0.f);   // Y2 = Y*T     (independent of Z2 below)
      mm64<0, true >(Z2, T, Z, 0.f, 0.f);   // Z2 = T*Z
      float* tp;
      tp = Y; Y = Y2; Y2 = tp;
      tp = Z; Z = Z2; Z2 = tp;
    }
  }

  // E = M - I into Z (reuse); ACC = (1/7)*I into T.
  for (int i = tid; i < 4096; i += 256) {
    const int r = i >> 6, c = i & 63;
    const float di = (r == c) ? 1.0f : 0.0f;
    Z[r * LDSTRIDE + c] = Y[r * LDSTRIDE + c] - di;
    T[r * LDSTRIDE + c] = di * (1.0f / 7.0f);
  }
  __syncthreads();

  // Horner for log(I+E) = E*(1 + E*(-1/2 + E*(1/3 + ...))) up to degree 7.
  float* ACC = T;
  float* NXT = Y2;
#pragma unroll 1
  for (int j = 0; j < 6; ++j) {
    const float cj = (j == 0) ? (-1.f / 6.f) : (j == 1) ? (1.f / 5.f)
                   : (j == 2) ? (-1.f / 4.f) : (j == 3) ? (1.f / 3.f)
                   : (j == 4) ? (-1.f / 2.f) : 1.f;
    mm64<2, true>(NXT, Z, ACC, cj, 0.f);    // NXT = cj*I + E*ACC
    float* tp = ACC; ACC = NXT; NXT = tp;
  }

  // out = 2^6 * (E*ACC) + log(c)*I  -> global
  const float logc = scal[1];
  mm64<3, false>(O, Z, ACC, 64.0f, logc);
}

extern "C" void kernel_launch(void* const* d_in, const int* in_sizes, int n_in,
                              void* d_out, int out_size, void* d_ws, size_t ws_size,
                              hipStream_t stream) {
  (void)n_in; (void)out_size; (void)d_ws; (void)ws_size;
  const float* X = (const float*)d_in[0];
  float* out = (float*)d_out;
  const int nb = in_sizes[0] / 4096;  // B = 8192 matrices of 64*64
  hipLaunchKernelGGL(logeig64_kernel, dim3(nb), dim3(256), 0, stream, X, out);
}